// ScaledDotProductAttention_12111807775438
// MI455X (gfx1250) — compile-verified
//
#include <hip/hip_runtime.h>

// Problem constants (from the reference): B=4, H=16, S=2048, D=128, fp32, causal.
constexpr int Bc = 4;
constexpr int Hc = 16;
constexpr int Sc = 2048;
constexpr int Dc = 128;

constexpr int ROWS   = 32;          // query rows per workgroup (2 row-tiles of 16)
constexpr int PSTR   = Sc + 4;      // LDS row stride (2052): conflict-free column float2 reads
constexpr int NTHR   = 256;         // 8 wave32
constexpr float QK_SCALE = 0.08838834764831845f;  // 1/sqrt(128)
constexpr size_t SMEM_BYTES = (size_t)ROWS * PSTR * sizeof(float);  // 262,656 B < 320 KB

typedef float v2f __attribute__((ext_vector_type(2)));
typedef float v8f __attribute__((ext_vector_type(8)));

#define WMMA_F32_4(A, B, C) \
    __builtin_amdgcn_wmma_f32_16x16x4_f32(false, (A), false, (B), (short)0, (C), false, false)

__global__ __launch_bounds__(NTHR, 1)
void sdpa_fused_fp32_wmma(const float* __restrict__ Q,
                          const float* __restrict__ K,
                          const float* __restrict__ V,
                          float* __restrict__ O,
                          float* __restrict__ W)
{
    const int tid  = threadIdx.x;
    const int wave = tid >> 5;
    const int lane = tid & 31;
    const int m16  = lane & 15;
    const bool hiHalf = (lane >= 16);
    const int cOff = hiHalf ? 2 : 0;   // K-offset within a 4-wide WMMA step for this lane
    const int hi8  = hiHalf ? 8 : 0;   // C/D row offset for hi half-wave

    const int bh      = blockIdx.y;
    const int rowBase = blockIdx.x * ROWS;

    const float* __restrict__ Qb = Q + (size_t)bh * Sc * Dc;
    const float* __restrict__ Kb = K + (size_t)bh * Sc * Dc;
    const float* __restrict__ Vb = V + (size_t)bh * Sc * Dc;
    float* __restrict__ Ob = O + (size_t)bh * Sc * Dc;
    float* __restrict__ Wb = W + (size_t)bh * Sc * Sc;

    extern __shared__ float Pt[];          // ROWS x PSTR fp32 logits/probs block
    __shared__ float s_m[ROWS];
    __shared__ float s_l[ROWS];

    // ============ Phase 1: S = Q K^T * scale (+causal -1e9), into LDS ============
    // Both row-tiles' A fragments live in registers so each K B-fragment feeds
    // two WMMAs (1 global b64 load : 2 v_wmma).
    {
        const int qmax0 = rowBase + 15;

        const float* qp0 = Qb + (size_t)(rowBase + m16) * Dc + cOff;
        const float* qp1 = Qb + (size_t)(rowBase + 16 + m16) * Dc + cOff;
        v2f afr0[32], afr1[32];
        #pragma unroll
        for (int t = 0; t < 32; ++t) afr0[t] = *(const v2f*)(qp0 + 4 * t);
        #pragma unroll
        for (int t = 0; t < 32; ++t) afr1[t] = *(const v2f*)(qp1 + 4 * t);

        const int ctEnd = (rowBase >> 4) + 2;   // tiles with n0 <= rowBase+31 (causal)
        for (int ct = wave; ct < ctEnd; ct += 8) {
            const int n0   = ct * 16;
            const int ncol = n0 + m16;
            const float* kp = Kb + (size_t)ncol * Dc + cOff;

            if (n0 <= qmax0) {
                // Both row-tiles active; tile1 is never diagonal here.
                v8f acc0 = {0.f,0.f,0.f,0.f,0.f,0.f,0.f,0.f};
                v8f acc1 = {0.f,0.f,0.f,0.f,0.f,0.f,0.f,0.f};
                #pragma unroll
                for (int t = 0; t < 32; ++t) {
                    v2f bf = *(const v2f*)(kp + 4 * t);
                    acc0 = WMMA_F32_4(afr0[t], bf, acc0);
                    acc1 = WMMA_F32_4(afr1[t], bf, acc1);
                }
                #pragma unroll
                for (int j = 0; j < 8; ++j) {
                    const int qg = rowBase + j + hi8;
                    float v0 = acc0[j] * QK_SCALE;
                    if (ncol > qg) v0 += -1.0e9f;          // diagonal tile of row-tile 0
                    Pt[(j + hi8) * PSTR + ncol] = v0;
                    Pt[(16 + j + hi8) * PSTR + ncol] = acc1[j] * QK_SCALE;
                }
            } else {
                // Only row-tile 1, and it's its diagonal tile (n0 == rowBase+16).
                v8f acc0 = {0.f,0.f,0.f,0.f,0.f,0.f,0.f,0.f};
                v8f acc1 = {0.f,0.f,0.f,0.f,0.f,0.f,0.f,0.f};
                #pragma unroll
                for (int t = 0; t < 16; ++t) {
                    v2f bfa = *(const v2f*)(kp + 8 * t);
                    v2f bfb = *(const v2f*)(kp + 8 * t + 4);
                    acc0 = WMMA_F32_4(afr1[2 * t], bfa, acc0);
                    acc1 = WMMA_F32_4(afr1[2 * t + 1], bfb, acc1);
                }
                #pragma unroll
                for (int j = 0; j < 8; ++j) {
                    const int qg = rowBase + 16 + j + hi8;
                    float v1 = (acc0[j] + acc1[j]) * QK_SCALE;
                    if (ncol > qg) v1 += -1.0e9f;
                    Pt[(16 + j + hi8) * PSTR + ncol] = v1;
                }
            }
        }
        // Fully-masked tiles are never written: phases 2/3 only touch n <= q.
    }
    __syncthreads();

    // ============ Phase 2: per-row online max/sum over the causal range ============
    for (int i = 0; i < 4; ++i) {
        const int r = wave * 4 + i;
        const int limit = rowBase + r + 1;        // valid keys for this query row
        float m = -3.0e38f;
        float l = 0.0f;
        for (int n = lane; n < limit; n += 32) {
            const float v = Pt[r * PSTR + n];
            const float mn = fmaxf(m, v);
            l = l * __expf(m - mn) + __expf(v - mn);
            m = mn;
        }
        #pragma unroll
        for (int off = 16; off > 0; off >>= 1) {
            const float mo = __shfl_xor(m, off, 32);
            const float lo = __shfl_xor(l, off, 32);
            const float mn = fmaxf(m, mo);
            l = l * __expf(m - mn) + lo * __expf(mo - mn);
            m = mn;
        }
        if (lane == 0) { s_m[r] = m; s_l[r] = l; }
    }
    __syncthreads();

    // ============ Phase 3: normalize in LDS + non-temporal weights stream ============
    // W is written once and never re-read: NT stores keep the 192MB L2 for K/V reuse.
    for (int r = 0; r < ROWS; ++r) {
        const float mm = s_m[r];
        const float il = 1.0f / s_l[r];
        const int rowLimit = rowBase + r + 1;
        float* wrow = Wb + (size_t)(rowBase + r) * Sc;
        for (int n = tid; n < Sc; n += NTHR) {
            float p = 0.0f;
            if (n < rowLimit)                    // wave-coherent branch: skips exp for masked tail
                p = __expf(Pt[r * PSTR + n] - mm) * il;
            Pt[r * PSTR + n] = p;
            __builtin_nontemporal_store(p, wrow + n);
        }
    }
    __syncthreads();

    // ============ Phase 4: O = P V (causally trimmed; A-frag shared by 2 tiles) ============
    {
        const int rt  = wave >> 2;               // waves 0-3 -> row-tile 0, 4-7 -> row-tile 1
        const int d0a = (wave & 3) * 32;         // two 16-wide d-tiles per wave
        const int d0b = d0a + 16;
        const int kMax = rowBase + rt * 16 + 16; // keys beyond q_max have P == 0

        v8f accA0 = {0.f,0.f,0.f,0.f,0.f,0.f,0.f,0.f};
        v8f accA1 = {0.f,0.f,0.f,0.f,0.f,0.f,0.f,0.f};
        v8f accB0 = {0.f,0.f,0.f,0.f,0.f,0.f,0.f,0.f};
        v8f accB1 = {0.f,0.f,0.f,0.f,0.f,0.f,0.f,0.f};
        const float* vpa = Vb + (size_t)(d0a + m16);
        const float* vpb = Vb + (size_t)(d0b + m16);
        const float* prow = &Pt[(rt * 16 + m16) * PSTR];

        for (int kk = 0; kk < kMax; kk += 8) {
            const int k0 = kk + cOff;
            const int k1 = kk + 4 + cOff;
            v2f a0 = *(const v2f*)(prow + k0);   // shared A-fragment (P) for both d-tiles
            v2f a1 = *(const v2f*)(prow + k1);
            v2f bA0; bA0.x = vpa[(size_t)k0 * Dc]; bA0.y = vpa[(size_t)(k0 + 1) * Dc];
            v2f bB0; bB0.x = vpb[(size_t)k0 * Dc]; bB0.y = vpb[(size_t)(k0 + 1) * Dc];
            v2f bA1; bA1.x = vpa[(size_t)k1 * Dc]; bA1.y = vpa[(size_t)(k1 + 1) * Dc];
            v2f bB1; bB1.x = vpb[(size_t)k1 * Dc]; bB1.y = vpb[(size_t)(k1 + 1) * Dc];
            accA0 = WMMA_F32_4(a0, bA0, accA0);
            accB0 = WMMA_F32_4(a0, bB0, accB0);
            accA1 = WMMA_F32_4(a1, bA1, accA1);
            accB1 = WMMA_F32_4(a1, bB1, accB1);
        }

        #pragma unroll
        for (int j = 0; j < 8; ++j) {
            const int qg = rowBase + rt * 16 + j + hi8;
            __builtin_nontemporal_store(accA0[j] + accA1[j], Ob + (size_t)qg * Dc + d0a + m16);
            __builtin_nontemporal_store(accB0[j] + accB1[j], Ob + (size_t)qg * Dc + d0b + m16);
        }
    }
}

extern "C" void kernel_launch(void* const* d_in, const int* in_sizes, int n_in,
                              void* d_out, int out_size, void* d_ws, size_t ws_size,
                              hipStream_t stream) {
    (void)in_sizes; (void)n_in; (void)d_ws; (void)ws_size; (void)out_size;

    const float* q = (const float*)d_in[0];
    const float* k = (const float*)d_in[1];
    const float* v = (const float*)d_in[2];
    // d_in[3] is the causal mask; causality is applied analytically in-kernel.

    float* out_o = (float*)d_out;                                  // [B,H,S,D]
    float* out_w = out_o + (size_t)Bc * Hc * Sc * Dc;              // [B,H,S,S]

    // Opt in to >64KB dynamic LDS (CDNA5 WGP has 320 KB).
    (void)hipFuncSetAttribute(reinterpret_cast<const void*>(sdpa_fused_fp32_wmma),
                              hipFuncAttributeMaxDynamicSharedMemorySize,
                              (int)SMEM_BYTES);

    dim3 grid(Sc / ROWS, Bc * Hc);   // 64 x 64
    dim3 block(NTHR);                // 8 wave32
    sdpa_fused_fp32_wmma<<<grid, block, SMEM_BYTES, stream>>>(q, k, v, out_o, out_w);
}